// SVDObserver_21285857919538
// MI455X (gfx1250) — compile-verified
//
#include <hip/hip_runtime.h>
#include <hip/hip_bf16.h>

typedef float v2f __attribute__((ext_vector_type(2)));
typedef float v8f __attribute__((ext_vector_type(8)));

#define CDIM 512
#define NPIX 4096
#define BATCH 64
#define KD 24
#define BLOCKS_PER_BATCH 8
#define WAVES_PER_BLOCK 4
#define PX_PER_WAVE (NPIX / (BLOCKS_PER_BATCH * WAVES_PER_BLOCK)) /* 128 */
#define CHUNKS_PER_WAVE (PX_PER_WAVE / 16)                        /* 8 */

#define WSTRIDE 517  /* 517 % 64 == 5 -> conflict-free strided LDS reads */
#define HSTRIDE 17
#define GSTR 33

/* output layout: S | Vh | features | novelty (concatenated flat) */
#define S_OFF 0
#define VH_OFF (BATCH * KD)                /* 1536 */
#define F_OFF (VH_OFF + BATCH * KD * KD)   /* 38400 */
#define NV_OFF (F_OFF + BATCH * 50)        /* 41600 */

static __device__ __forceinline__ bool finitef(float v) {
    return __builtin_isfinite(v);
}

/* ---------------- Kernel 1: fused projection + Gram via fp32 WMMA -------- */
__global__ __launch_bounds__(128) void proj_gram_kernel(
    const float* __restrict__ x, const float* __restrict__ w,
    float* __restrict__ Gpart)
{
    __shared__ float wLds[32 * WSTRIDE];                       /* w padded to 32 rows */
    __shared__ float hbuf[WAVES_PER_BLOCK * 32 * HSTRIDE];     /* per-wave h^T tile   */
    __shared__ float Gsh[WAVES_PER_BLOCK * 32 * GSTR];         /* per-wave partial G  */

    const int tid   = threadIdx.x;
    const int lane  = tid & 31;
    const int wav   = tid >> 5;
    const int b     = blockIdx.x >> 3;
    const int blk   = blockIdx.x & 7;
    const int l15   = lane & 15;
    const int chalf = (lane >> 4) * 2;  /* K-pair owned by this half-wave */
    const int half8 = (lane >> 4) * 8;  /* D-layout row offset            */

    /* stage w^pad[j][c] into LDS (rows 24..31 zero); coalesced global reads */
    for (int i = tid; i < 32 * CDIM; i += 128) {
        int j = i >> 9, c = i & 511;
        wLds[j * WSTRIDE + c] = (j < KD) ? w[j * CDIM + c] : 0.0f;
    }
    __syncthreads();

    const float* xb = x + (size_t)b * CDIM * NPIX;
    float* hb = &hbuf[wav * 32 * HSTRIDE];

    v8f g00 = {}; v8f g01 = {}; v8f g11 = {};

    const int pxw = (blk * WAVES_PER_BLOCK + wav) * PX_PER_WAVE;
    for (int ch = 0; ch < CHUNKS_PER_WAVE; ++ch) {
        const int px0 = pxw + ch * 16;
        const float* xp = xb + px0 + l15;

        /* h^T(16k x 16px) tiles: A = w-tile (16k x 4c), B = x (4c x 16px) */
        v8f h0 = {}; v8f h1 = {};
        #pragma unroll 4
        for (int c0 = 0; c0 < CDIM; c0 += 4) {
            const int cc = c0 + chalf;
            v2f bx;
            bx.x = xp[(size_t)cc * NPIX];
            bx.y = xp[(size_t)(cc + 1) * NPIX];
            v2f a0, a1;
            a0.x = wLds[l15 * WSTRIDE + cc];
            a0.y = wLds[l15 * WSTRIDE + cc + 1];
            a1.x = wLds[(16 + l15) * WSTRIDE + cc];
            a1.y = wLds[(16 + l15) * WSTRIDE + cc + 1];
            h0 = __builtin_amdgcn_wmma_f32_16x16x4_f32(false, a0, false, bx,
                                                       (short)0, h0, false, false);
            h1 = __builtin_amdgcn_wmma_f32_16x16x4_f32(false, a1, false, bx,
                                                       (short)0, h1, false, false);
        }

        /* spill h^T tile: element (k = tile*16 + half8 + r, px = l15) */
        #pragma unroll
        for (int r = 0; r < 8; ++r) {
            hb[(half8 + r) * HSTRIDE + l15]      = h0[r];
            hb[(16 + half8 + r) * HSTRIDE + l15] = h1[r];
        }
        __builtin_amdgcn_wave_barrier();

        /* Gram: G += h^T * h ; A-frag and B-frag share the same LDS pattern */
        #pragma unroll
        for (int p0 = 0; p0 < 16; p0 += 4) {
            const int pp = p0 + chalf;
            v2f f0, f1;
            f0.x = hb[l15 * HSTRIDE + pp];
            f0.y = hb[l15 * HSTRIDE + pp + 1];
            f1.x = hb[(16 + l15) * HSTRIDE + pp];
            f1.y = hb[(16 + l15) * HSTRIDE + pp + 1];
            g00 = __builtin_amdgcn_wmma_f32_16x16x4_f32(false, f0, false, f0,
                                                        (short)0, g00, false, false);
            g01 = __builtin_amdgcn_wmma_f32_16x16x4_f32(false, f0, false, f1,
                                                        (short)0, g01, false, false);
            g11 = __builtin_amdgcn_wmma_f32_16x16x4_f32(false, f1, false, f1,
                                                        (short)0, g11, false, false);
        }
        __builtin_amdgcn_wave_barrier();
    }

    /* each wave writes its full 32x32 partial (tile 10 mirrored from 01) */
    float* Gw = &Gsh[wav * 32 * GSTR];
    #pragma unroll
    for (int r = 0; r < 8; ++r) {
        const int row = half8 + r;
        Gw[row * GSTR + l15]             = g00[r];
        Gw[row * GSTR + 16 + l15]        = g01[r];
        Gw[(16 + l15) * GSTR + row]      = g01[r];  /* symmetric mirror */
        Gw[(16 + row) * GSTR + 16 + l15] = g11[r];
    }
    __syncthreads();

    /* deterministic in-block reduction, then plain store of the partial */
    if (wav == 0) {
        float* dst = Gpart + (size_t)blockIdx.x * 1024;
        for (int i = 0; i < 32; ++i) {
            float v = Gsh[0 * 32 * GSTR + i * GSTR + lane]
                    + Gsh[1 * 32 * GSTR + i * GSTR + lane]
                    + Gsh[2 * 32 * GSTR + i * GSTR + lane]
                    + Gsh[3 * 32 * GSTR + i * GSTR + lane];
            dst[i * 32 + lane] = v;
        }
    }
}

/* ---------------- Kernel 2: fixed-order cross-block reduction ------------ */
__global__ __launch_bounds__(256) void reduce_g_kernel(
    const float* __restrict__ Gpart, float* __restrict__ Gfin)
{
    int idx = blockIdx.x * 256 + threadIdx.x;
    if (idx >= BATCH * 1024) return;
    int b = idx >> 10;
    int e = idx & 1023;
    const float* p = Gpart + (size_t)b * BLOCKS_PER_BATCH * 1024 + e;
    float v = 0.0f;
    #pragma unroll
    for (int k = 0; k < BLOCKS_PER_BATCH; ++k) v += p[k * 1024];
    Gfin[idx] = v;
}

/* ---------------- Kernel 3: parallel Jacobi eigh + features -------------- */
__global__ __launch_bounds__(32) void eigh_feat_kernel(
    const float* __restrict__ G, const float* __restrict__ ema,
    float* __restrict__ out)
{
    __shared__ float A[KD * 25];
    __shared__ float V[KD * 25];
    __shared__ float csb[24];
    __shared__ int   prs[24];
    __shared__ float Ssh[KD];
    __shared__ float Dsh[KD];
    __shared__ float red[32];

    const int b = blockIdx.x;
    const int lane = threadIdx.x;

    for (int i = lane; i < KD * KD; i += 32) {
        int r = i / KD, c = i - r * KD;
        A[r * 25 + c] = G[(size_t)b * 1024 + r * 32 + c];
        V[r * 25 + c] = (r == c) ? 1.0f : 0.0f;
    }
    __syncthreads();

    /* tournament-ordered parallel Jacobi: 12 disjoint rotations per round */
    for (int sw = 0; sw < 10; ++sw) {
        for (int rr = 0; rr < 23; ++rr) {
            if (lane < 12) {
                int i0 = lane, i1 = 23 - lane;
                int p = (i0 == 0) ? 0 : ((i0 - 1 + rr) % 23) + 1;
                int q = ((i1 - 1 + rr) % 23) + 1;
                if (p > q) { int t = p; p = q; q = t; }
                float app = A[p * 25 + p];
                float aqq = A[q * 25 + q];
                float apq = A[p * 25 + q];
                float c = 1.0f, s = 0.0f;
                if (apq != 0.0f) {
                    float tau = (aqq - app) / (2.0f * apq);
                    float tt = copysignf(1.0f, tau) /
                               (fabsf(tau) + sqrtf(1.0f + tau * tau));
                    c = 1.0f / sqrtf(1.0f + tt * tt);
                    s = tt * c;
                }
                csb[lane * 2] = c; csb[lane * 2 + 1] = s;
                prs[lane * 2] = p; prs[lane * 2 + 1] = q;
            }
            __syncthreads();
            /* row phase: A <- J^T A   (288 disjoint element-pairs) */
            for (int wdx = 0; wdx < 9; ++wdx) {
                int idx = lane + 32 * wdx;
                int m = idx / 24, j = idx - m * 24;
                int p = prs[m * 2], q = prs[m * 2 + 1];
                float c = csb[m * 2], s = csb[m * 2 + 1];
                float ap = A[p * 25 + j], aq = A[q * 25 + j];
                A[p * 25 + j] = c * ap - s * aq;
                A[q * 25 + j] = s * ap + c * aq;
            }
            __syncthreads();
            /* col phase: A <- A J ; V <- V J */
            for (int wdx = 0; wdx < 9; ++wdx) {
                int idx = lane + 32 * wdx;
                int m = idx / 24, i = idx - m * 24;
                int p = prs[m * 2], q = prs[m * 2 + 1];
                float c = csb[m * 2], s = csb[m * 2 + 1];
                float aip = A[i * 25 + p], aiq = A[i * 25 + q];
                A[i * 25 + p] = c * aip - s * aiq;
                A[i * 25 + q] = s * aip + c * aiq;
                float vip = V[i * 25 + p], viq = V[i * 25 + q];
                V[i * 25 + p] = c * vip - s * viq;
                V[i * 25 + q] = s * vip + c * viq;
            }
            __syncthreads();
        }
    }

    /* rank-sort eigenvalues descending; write S, Vh, novelty */
    float Sval = 0.0f; int rank = 0;
    if (lane < KD) {
        float ev = A[lane * 25 + lane];
        for (int j = 0; j < KD; ++j) {
            float evj = A[j * 25 + j];
            if ((evj > ev) || (evj == ev && j < lane)) rank += 1;
        }
        float sq = sqrtf(fmaxf(ev, 0.0f));
        Sval = fmaxf(sq, 1e-6f);
        if (!finitef(Sval)) Sval = 1.0f;
        Ssh[rank] = Sval;

        float dsum = 0.0f;
        for (int cI = 0; cI < KD; ++cI) {
            float vv = V[cI * 25 + lane];     /* eigvec = column of V */
            if (!finitef(vv)) vv = 0.0f;
            out[VH_OFF + (size_t)b * (KD * KD) + rank * KD + cI] = vv;
            dsum += vv * vv;
        }
        red[lane] = dsum;
        float d = V[rank * 25 + lane];        /* Vh[rank][rank] */
        if (!finitef(d)) d = 0.0f;
        Dsh[rank] = d;

        out[S_OFF + b * KD + rank]  = Sval;
        out[NV_OFF + b * KD + rank] = Sval - ema[rank];
    } else {
        red[lane] = 0.0f;
    }
    __syncthreads();

    float tot = 0.0f, dg = 0.0f, ssum = 0.0f;
    for (int j = 0; j < KD; ++j) {
        tot += red[j];
        float dj = Dsh[j]; dg += dj * dj;
        ssum += Ssh[j];
    }
    float inv = 1.0f / (ssum + 1e-8f);
    float ent = 0.0f;
    for (int j = 0; j < KD; ++j) {
        float sn = Ssh[j] * inv;
        ent -= sn * logf(fmaxf(sn, 1e-8f));
    }
    if (lane < KD) {
        float sn = Ssh[lane] * inv;
        if (!finitef(sn)) sn = 0.0f;
        out[F_OFF + b * 50 + lane] = sn;
        out[F_OFF + b * 50 + KD + lane] = Dsh[lane];
    }
    if (lane == 0) {
        float off = fmaxf(tot - dg, 0.0f);
        if (!finitef(off)) off = 0.0f;
        if (!finitef(ent)) ent = 0.0f;
        out[F_OFF + b * 50 + 48] = off;
        out[F_OFF + b * 50 + 49] = ent;
    }
}

/* ------------------------------------------------------------------------ */
extern "C" void kernel_launch(void* const* d_in, const int* in_sizes, int n_in,
                              void* d_out, int out_size, void* d_ws, size_t ws_size,
                              hipStream_t stream)
{
    const float* x   = (const float*)d_in[0];
    const float* w   = (const float*)d_in[1];
    const float* ema = (const float*)d_in[2];
    float* out = (float*)d_out;

    float* Gpart = (float*)d_ws;                                     /* 64*8*1024 f32 */
    float* Gfin  = Gpart + (size_t)BATCH * BLOCKS_PER_BATCH * 1024;  /* 64*1024 f32  */

    proj_gram_kernel<<<BATCH * BLOCKS_PER_BATCH, 128, 0, stream>>>(x, w, Gpart);
    reduce_g_kernel<<<(BATCH * 1024 + 255) / 256, 256, 0, stream>>>(Gpart, Gfin);
    eigh_feat_kernel<<<BATCH, 32, 0, stream>>>(Gfin, ema, out);
}